// AdversarialGAT_11184094839375
// MI455X (gfx1250) — compile-verified
//
#include <hip/hip_runtime.h>
#include <math.h>

typedef float v2f __attribute__((ext_vector_type(2)));
typedef float v8f __attribute__((ext_vector_type(8)));

#define NN 100000
#define EE 1600000
#define GG 256
#define DD 32
#define HH 100
#define FF 32

// ---------------------------------------------------------------- zero
__global__ void k_zero(float* __restrict__ p, int n) {
    int i = blockIdx.x * blockDim.x + threadIdx.x;
    if (i < n) p[i] = 0.0f;
}

// ---------------------------------------------------------------- degrees
__global__ void k_degrees(const int* __restrict__ src, const int* __restrict__ dst,
                          float* __restrict__ outdeg, float* __restrict__ indeg) {
    int e = blockIdx.x * blockDim.x + threadIdx.x;
    if (e < EE) {
        atomicAdd(&outdeg[src[e]], 1.0f);
        atomicAdd(&indeg[dst[e]], 1.0f);
    }
}

// ---------------------------------------------------------------- deg -> scale (in place)
__global__ void k_scales(float* __restrict__ oscale, float* __restrict__ iscale,
                         const float* __restrict__ mask) {
    int i = blockIdx.x * blockDim.x + threadIdx.x;
    if (i < NN) {
        float od = oscale[i]; od = od < 1.0f ? 1.0f : od;
        float s = rsqrtf(od);
        if (mask) s *= mask[i];
        oscale[i] = s;
        float id = iscale[i]; id = id < 1.0f ? 1.0f : id;
        iscale[i] = rsqrtf(id);
    }
}

// ---------------------------------------------------------------- edge scatter-add
// 16 lanes per edge, 2 features per lane: b64 embed loads, coalesced 128B rows,
// half the redundant per-edge scalar loads vs lane-per-feature.
__global__ void k_scatter(const int* __restrict__ src, const int* __restrict__ dst,
                          const int* __restrict__ tokens, const float* __restrict__ embed,
                          const float* __restrict__ srcscale, float* __restrict__ agg) {
    int t = blockIdx.x * blockDim.x + threadIdx.x;
    int e = t >> 4;
    int d = (t & 15) * 2;
    if (e < EE) {
        int s   = src[e];
        int tok = tokens[s];
        float sc = srcscale[s];
        float2 ev = *(const float2*)(embed + tok * DD + d);
        float* ap = agg + dst[e] * DD + d;
        atomicAdd(ap,     ev.x * sc);
        atomicAdd(ap + 1, ev.y * sc);
    }
}

// ---------------------------------------------------------------- GEMM (agg*inscale @ W + b, relu) fused with segment-max pool
// one wave computes one 16x16 tile of h; 8x WMMA_F32_16X16X4_F32 over K=32.
// Padded B columns (colg >= 100) are NOT zeroed: B column j only feeds D column j,
// and the epilogue never stores those lanes. Clamp keeps loads in-bounds.
__global__ void k_gemm_pool(const float* __restrict__ agg, const float* __restrict__ iscale,
                            const float* __restrict__ W, const float* __restrict__ bias,
                            const int* __restrict__ gid, float* __restrict__ pooled) {
    const int rowTiles = NN / 16;          // 6250 (exact)
    const int colTiles = (HH + 15) / 16;   // 7 (H padded 100->112)
    int wave = (blockIdx.x * blockDim.x + threadIdx.x) >> 5;
    int lane = threadIdx.x & 31;
    if (wave >= rowTiles * colTiles) return;   // wave-uniform: EXEC stays all-1 for WMMA
    int rowTile = wave / colTiles;
    int colTile = wave % colTiles;
    int row  = lane & 15;          // A row (M) for this lane, also B/D column index
    int kh   = lane >> 4;          // K-half select
    int node = rowTile * 16 + row;
    int colg = colTile * 16 + row;
    int colc = colg < HH ? colg : HH - 1;   // in-bounds clamp, no exec masking
    float s = iscale[node];
    const float* arow = agg + node * DD;
    const float* wcol = W + colc;

    v8f c = {};
#pragma unroll
    for (int ch = 0; ch < 8; ++ch) {
        int kb = ch * 4 + kh * 2;
        v2f a;
        a.x = arow[kb]     * s;
        a.y = arow[kb + 1] * s;
        v2f b;
        b.x = wcol[kb * HH];
        b.y = wcol[(kb + 1) * HH];
        c = __builtin_amdgcn_wmma_f32_16x16x4_f32(false, a, false, b,
                                                  (short)0, c, false, false);
    }

    if (colg < HH) {
        float bc = bias[colg];
#pragma unroll
        for (int r = 0; r < 8; ++r) {
            int nrow = rowTile * 16 + r + kh * 8;   // D layout: vgpr r, lanes16-31 -> M=r+8
            float h = c[r] + bc;
            h = h > 0.0f ? h : 0.0f;                // relu => h >= 0, u32 order == float order
            int g = gid[nrow];
            atomicMax((unsigned int*)&pooled[g * HH + colg], __float_as_uint(h));
        }
    }
}

// ---------------------------------------------------------------- FC: pooled(256x100) @ fc_W(100x32) + fc_b
__global__ void k_fc(const float* __restrict__ pooled, const float* __restrict__ W,
                     const float* __restrict__ bias, float* __restrict__ out) {
    const int rowTiles = GG / 16;   // 16
    const int colTiles = FF / 16;   // 2
    int wave = (blockIdx.x * blockDim.x + threadIdx.x) >> 5;
    int lane = threadIdx.x & 31;
    if (wave >= rowTiles * colTiles) return;
    int rowTile = wave / colTiles;
    int colTile = wave % colTiles;
    int row  = lane & 15;
    int kh   = lane >> 4;
    int grow = rowTile * 16 + row;
    int gcol = colTile * 16 + row;

    v8f c = {};
#pragma unroll
    for (int ch = 0; ch < 25; ++ch) {              // K = 100 = 25 * 4 (exact)
        int kb = ch * 4 + kh * 2;
        v2f a;
        a.x = pooled[grow * HH + kb];
        a.y = pooled[grow * HH + kb + 1];
        v2f b;
        b.x = W[kb * FF + gcol];
        b.y = W[(kb + 1) * FF + gcol];
        c = __builtin_amdgcn_wmma_f32_16x16x4_f32(false, a, false, b,
                                                  (short)0, c, false, false);
    }
    float bb = bias[gcol];
#pragma unroll
    for (int r = 0; r < 8; ++r) {
        int orow = rowTile * 16 + r + kh * 8;
        out[orow * FF + gcol] = c[r] + bb;
    }
}

// ---------------------------------------------------------------- defect head: sigmoid(xb @ dW + db)
__global__ void k_defect(const float* __restrict__ xb, const float* __restrict__ dW,
                         const float* __restrict__ db, float* __restrict__ out) {
    int r = blockIdx.x * blockDim.x + threadIdx.x;
    if (r < GG) {
        float acc = db[0];
#pragma unroll
        for (int k = 0; k < FF; ++k) acc += xb[r * FF + k] * dW[k];
        out[r] = 1.0f / (1.0f + expf(-acc));
    }
}

// ================================================================ launch
extern "C" void kernel_launch(void* const* d_in, const int* in_sizes, int n_in,
                              void* d_out, int out_size, void* d_ws, size_t ws_size,
                              hipStream_t stream) {
    (void)in_sizes; (void)n_in; (void)out_size; (void)ws_size;

    const int* tokens[3] = { (const int*)d_in[0], (const int*)d_in[4], (const int*)d_in[8]  };
    const int* src[3]    = { (const int*)d_in[1], (const int*)d_in[5], (const int*)d_in[9]  };
    const int* dst[3]    = { (const int*)d_in[2], (const int*)d_in[6], (const int*)d_in[10] };
    const int* gid[3]    = { (const int*)d_in[3], (const int*)d_in[7], (const int*)d_in[11] };
    const float* mask    = (const float*)d_in[12];
    const float* embed   = (const float*)d_in[13];
    const float* gcn_W   = (const float*)d_in[14];
    const float* gcn_b   = (const float*)d_in[15];
    const float* fc_W    = (const float*)d_in[16];
    const float* fc_b    = (const float*)d_in[17];
    const float* def_W   = (const float*)d_in[18];
    const float* def_b   = (const float*)d_in[19];
    float* out = (float*)d_out;

    // workspace layout (bytes, 128B-aligned regions)
    char* ws = (char*)d_ws;
    const size_t SC_BYTES  = ((size_t)NN * 4 + 127) & ~(size_t)127;
    float* oscale = (float*)(ws);
    float* iscale = (float*)(ws + SC_BYTES);
    float* agg    = (float*)(ws + 2 * SC_BYTES);
    const size_t AGG_BYTES = (size_t)NN * DD * 4;                          // 12.8 MB
    float* pooled = (float*)(ws + 2 * SC_BYTES + AGG_BYTES);
    const int zeroFloats = (int)((2 * SC_BYTES + AGG_BYTES) / 4) + GG * HH;

    // output flat-concat offsets: x_batch_mmd | defect_prob | x_src_mmd | x_tar_mmd
    float* mmd_out[3]  = { out, out + GG * FF + GG, out + GG * FF + GG + GG * FF };
    float* defect_out  = out + GG * FF;

    const int T = 256;
    for (int v = 0; v < 3; ++v) {
        k_zero<<<(zeroFloats + T - 1) / T, T, 0, stream>>>((float*)ws, zeroFloats);
        k_degrees<<<(EE + T - 1) / T, T, 0, stream>>>(src[v], dst[v], oscale, iscale);
        k_scales<<<(NN + T - 1) / T, T, 0, stream>>>(oscale, iscale, v == 0 ? mask : nullptr);
        k_scatter<<<(EE * 16) / T, T, 0, stream>>>(src[v], dst[v], tokens[v], embed, oscale, agg);
        {
            int waves = (NN / 16) * ((HH + 15) / 16);
            int thr   = waves * 32;
            k_gemm_pool<<<(thr + T - 1) / T, T, 0, stream>>>(agg, iscale, gcn_W, gcn_b,
                                                             gid[v], pooled);
        }
        {
            int waves = (GG / 16) * (FF / 16);
            int thr   = waves * 32;
            k_fc<<<(thr + T - 1) / T, T, 0, stream>>>(pooled, fc_W, fc_b, mmd_out[v]);
        }
        if (v == 0)
            k_defect<<<1, T, 0, stream>>>(mmd_out[0], def_W, def_b, defect_out);
    }
}